// PSJALoss_12867722019335
// MI455X (gfx1250) — compile-verified
//
#include <hip/hip_runtime.h>
#include <math.h>

#define BAGS 16
#define NPB  2048
#define DIM  1024
#define KH   15
#define KL   5
#define ROWS_H (BAGS * KH)          // 240
#define ROWS   (BAGS * (KH + KL))   // 320
#define INV_TAU (1.0f / 16.0f)

typedef __attribute__((ext_vector_type(2))) float v2f;
typedef __attribute__((ext_vector_type(8))) float v8f;

// ---------------- Kernel 0: global score stats (max, min, mean) ----------------
__global__ void k_stats(const float* __restrict__ score, float* __restrict__ stats) {
    __shared__ float smax[1024], smin[1024], ssum[1024];
    const int t = threadIdx.x;
    float mx = -1e30f, mn = 1e30f, sm = 0.0f;
    for (int i = t; i < BAGS * NPB; i += blockDim.x) {
        float v = score[i];
        mx = fmaxf(mx, v);
        mn = fminf(mn, v);
        sm += v;
    }
    smax[t] = mx; smin[t] = mn; ssum[t] = sm;
    __syncthreads();
    for (int s = blockDim.x >> 1; s > 0; s >>= 1) {
        if (t < s) {
            smax[t] = fmaxf(smax[t], smax[t + s]);
            smin[t] = fminf(smin[t], smin[t + s]);
            ssum[t] += ssum[t + s];
        }
        __syncthreads();
    }
    if (t == 0) {
        stats[0] = smax[0];                              // gmax
        stats[1] = smin[0];                              // gmin
        stats[2] = ssum[0] / (float)(BAGS * NPB);        // mean(score)
    }
}

// ---------------- Kernel 1: per-bag selection via stable pairwise ranks --------
// One block per bag. Rank within mask under stable (score, index) order replaces
// the reference's argsort + cumsum. Selected rows are gathered into bank.
__global__ void k_select(const float* __restrict__ fea,
                         const float* __restrict__ score,
                         const int*   __restrict__ label,
                         const float* __restrict__ stats,
                         float* __restrict__ bank,
                         int* __restrict__ valid,
                         int* __restrict__ lab) {
    const int b = blockIdx.x;
    const int t = threadIdx.x;
    __shared__ float ss[NPB];
    __shared__ int selH[KH];
    __shared__ int selL[KL];

    for (int i = t; i < NPB; i += blockDim.x) ss[i] = score[b * NPB + i];
    if (t < KH) selH[t] = -1;
    if (t < KL) selL[t] = -1;
    __syncthreads();

    const float gmax = stats[0];
    const float gmin = stats[1];
    const float hlo = gmax - 0.3f;
    const float lhi = gmin + 1e-9f;

    for (int i = t; i < NPB; i += blockDim.x) {
        const float s = ss[i];
        // high-score window: first KH in stable ascending order
        if (s >= hlo && s <= gmax) {
            int r = 0;
            for (int j = 0; j < NPB && r < KH; ++j) {
                const float sj = ss[j];
                if (sj >= hlo && sj <= gmax && (sj < s || (sj == s && j < i))) ++r;
            }
            if (r < KH) selH[r] = i;     // ranks are unique -> no conflicts
        }
        // low-score window: even ranks 0,2,...,8 -> slots 0..4
        if (s >= gmin && s <= lhi) {
            int r = 0;
            for (int j = 0; j < NPB && r < 2 * KL; ++j) {
                const float sj = ss[j];
                if (sj >= gmin && sj <= lhi && (sj < s || (sj == s && j < i))) ++r;
            }
            if ((r & 1) == 0 && (r >> 1) < KL) selL[r >> 1] = i;
        }
    }
    __syncthreads();

    const int lb = label[b];
    for (int k = 0; k < KH; ++k) {
        const int row = b * KH + k;
        const int src = selH[k];
        const float* srcp = fea + ((size_t)(b * NPB + src)) * DIM;
        for (int d = t; d < DIM; d += blockDim.x)
            bank[(size_t)row * DIM + d] = (src >= 0) ? srcp[d] : 0.0f;
        if (t == 0) { valid[row] = (src >= 0) ? 1 : 0; lab[row] = lb; }
    }
    for (int k = 0; k < KL; ++k) {
        const int row = ROWS_H + b * KL + k;
        const int src = selL[k];
        const float* srcp = fea + ((size_t)(b * NPB + src)) * DIM;
        for (int d = t; d < DIM; d += blockDim.x)
            bank[(size_t)row * DIM + d] = (src >= 0) ? srcp[d] : 0.0f;
        if (t == 0) { valid[row] = (src >= 0) ? 1 : 0; lab[row] = 2; } // N_CLASSES
    }
}

// ---------------- Kernel 2: Gram matrix S = bank * bank^T via fp32 WMMA --------
// One wave (32 lanes) per 16x16 output tile. V_WMMA_F32_16X16X4_F32.
// 32-bit A 16x4 layout: lanes 0-15 hold M=lane with K=0,1 in the two VGPRs,
// lanes 16-31 hold K=2,3. B 4x16 mirrors it with lanes holding N.
__global__ void k_gram(const float* __restrict__ bank, float* __restrict__ S) {
    const int i0 = blockIdx.x * 16;
    const int j0 = blockIdx.y * 16;
    const int lane = threadIdx.x;       // 0..31, full wave active (EXEC all 1s)
    const int m = lane & 15;
    const int koff = (lane >> 4) * 2;   // 0 for lanes 0-15, 2 for lanes 16-31

    const float* arow = bank + (size_t)(i0 + m) * DIM;
    const float* brow = bank + (size_t)(j0 + m) * DIM;

    v8f c = {};
    for (int k = 0; k < DIM; k += 4) {
        v2f a = *(const v2f*)(arow + k + koff);
        v2f bb = *(const v2f*)(brow + k + koff);
        c = __builtin_amdgcn_wmma_f32_16x16x4_f32(
            /*neg_a=*/false, a, /*neg_b=*/false, bb,
            /*c_mod=*/(short)0, c, /*reuse_a=*/false, /*reuse_b=*/false);
    }

    // C/D layout: VGPR r -> M = r (+8 for lanes 16-31), N = lane & 15
    const int row0 = i0 + ((lane >= 16) ? 8 : 0);
    const int col = j0 + m;
#pragma unroll
    for (int r = 0; r < 8; ++r)
        S[(size_t)(row0 + r) * ROWS + col] = c[r];
}

// ---------------- Kernel 3: per-row contrastive term ---------------------------
__global__ void k_rows(const float* __restrict__ S,
                       const int* __restrict__ valid,
                       const int* __restrict__ lab,
                       const int* __restrict__ label,
                       float* __restrict__ per) {
    const int r = blockIdx.x;
    const int t = threadIdx.x;
    __shared__ float sden[64], snum[64];
    const int qlab = (r < ROWS_H) ? label[r / KH] : 2;
    float den = 0.0f, num = 0.0f;
    for (int n = t; n < ROWS; n += blockDim.x) {
        if (valid[n]) {
            const float e = expf(S[(size_t)r * ROWS + n] * INV_TAU);
            den += e;
            if (lab[n] == qlab) num += e;
        }
    }
    sden[t] = den; snum[t] = num;
    __syncthreads();
    for (int s = blockDim.x >> 1; s > 0; s >>= 1) {
        if (t < s) { sden[t] += sden[t + s]; snum[t] += snum[t + s]; }
        __syncthreads();
    }
    if (t == 0)
        per[r] = valid[r] ? -logf(snum[0] / sden[0]) : 0.0f;
}

// ---------------- Kernel 4: finalize scalar loss -------------------------------
__global__ void k_final(const float* __restrict__ per,
                        const int* __restrict__ valid,
                        const float* __restrict__ stats,
                        const float* __restrict__ t_logit,
                        const float* __restrict__ ori,
                        const float* __restrict__ l2w,
                        float* __restrict__ out) {
    if (threadIdx.x == 0 && blockIdx.x == 0) {
        float acc = 0.0f;
        for (int b = 0; b < BAGS; ++b) {
            float s = 0.0f; int c = 0;
            for (int k = 0; k < KH; ++k) { int r = b * KH + k; s += per[r]; c += valid[r]; }
            for (int k = 0; k < KL; ++k) { int r = ROWS_H + b * KL + k; s += per[r]; c += valid[r]; }
            acc += s / (float)c;
        }
        const float contrastive = acc / (float)BAGS;

        const float t = t_logit[0];
        float ce = 0.0f;
        for (int b = 0; b < BAGS; ++b)
            ce += logf(expf(ori[b * 2 + 0]) + expf(ori[b * 2 + 1])) - t;
        ce /= (float)BAGS;

        const float Dm = stats[2] / stats[0];
        const float l2 = 1e-4f * (1.0f - Dm) * (1.0f - Dm) * l2w[0];
        out[0] = contrastive + ce + l2;
    }
}

extern "C" void kernel_launch(void* const* d_in, const int* in_sizes, int n_in,
                              void* d_out, int out_size, void* d_ws, size_t ws_size,
                              hipStream_t stream) {
    const float* fea     = (const float*)d_in[0];
    const float* score   = (const float*)d_in[1];
    const int*   label   = (const int*)  d_in[2];
    const float* t_logit = (const float*)d_in[3];
    const float* ori     = (const float*)d_in[4];
    const float* l2w     = (const float*)d_in[5];
    float* out = (float*)d_out;

    // Workspace layout (floats, then ints) ~1.73 MB total
    float* wsf   = (float*)d_ws;
    float* bank  = wsf;                        // ROWS*DIM
    float* S     = bank + (size_t)ROWS * DIM;  // ROWS*ROWS
    float* per   = S + (size_t)ROWS * ROWS;    // ROWS
    float* stats = per + ROWS;                 // 4
    int*   valid = (int*)(stats + 4);          // ROWS
    int*   lab   = valid + ROWS;               // ROWS

    k_stats <<<1, 1024, 0, stream>>>(score, stats);
    k_select<<<BAGS, 256, 0, stream>>>(fea, score, label, stats, bank, valid, lab);
    k_gram  <<<dim3(ROWS / 16, ROWS / 16), 32, 0, stream>>>(bank, S);
    k_rows  <<<ROWS, 64, 0, stream>>>(S, valid, lab, label, per);
    k_final <<<1, 32, 0, stream>>>(per, valid, stats, t_logit, ori, l2w, out);
}